// ContextualConv2d_77171972374874
// MI455X (gfx1250) — compile-verified
//
#include <hip/hip_runtime.h>

typedef float v2f __attribute__((ext_vector_type(2)));
typedef float v8f __attribute__((ext_vector_type(8)));

#define NB    32
#define CIN   128
#define HDIM  56
#define WDIM  56
#define HW    (HDIM * WDIM)      // 3136
#define COUT  256
#define GROUPS 4
#define CING  (CIN / GROUPS)     // 32
#define COG   (COUT / GROUPS)    // 64
#define KTOT  (CING * 9)         // 288
#define CDIM  64

// LDS x-tile: K-interleaved layout  sX[ci/4][row*58+col][ci%4]
// -> the (k0, k0+1) pair a lane-half needs is one aligned ds_load_b64.
#define TROWS 6
#define TCOLS 58
#define XROW  (TROWS * TCOLS)    // 348 cells per ci
#define XBLK  (XROW * 4)         // 1392 words per 4-ci block
// total: 8 blocks * 1392 * 4 B = 44544 B

typedef __attribute__((address_space(1))) int* gas_i32p;   // global
typedef __attribute__((address_space(3))) int* las_i32p;   // LDS

// ---------------------------------------------------------------------------
// CDNA5 async global->LDS copy (ASYNCcnt-tracked, no VGPR round-trip).
// ---------------------------------------------------------------------------
__device__ __forceinline__ void async_g2l_b32(float* lds_dst, const float* gsrc) {
#if __has_builtin(__builtin_amdgcn_global_load_async_to_lds_b32)
    __builtin_amdgcn_global_load_async_to_lds_b32(
        (gas_i32p)(size_t)gsrc,
        (las_i32p)(unsigned int)(size_t)lds_dst,
        0, 0);
#else
    unsigned int loff = (unsigned int)(size_t)lds_dst;
    asm volatile("global_load_async_to_lds_b32 %0, %1, off"
                 :: "v"(loff), "v"(gsrc) : "memory");
#endif
}

__device__ __forceinline__ void wait_async0() {
#if __has_builtin(__builtin_amdgcn_s_wait_asynccnt)
    __builtin_amdgcn_s_wait_asynccnt(0);
#else
    asm volatile("s_wait_asynccnt 0x0" ::: "memory");
#endif
}

// ---------------------------------------------------------------------------
// Kernel 1: ctx[n, co] = bias[co] + dot(c[n, :], c_weight[co, :])
// ---------------------------------------------------------------------------
__global__ void ctx_bias_kernel(const float* __restrict__ c,
                                const float* __restrict__ cw,
                                const float* __restrict__ bias,
                                float* __restrict__ ctx) {
    int idx = blockIdx.x * blockDim.x + threadIdx.x;
    if (idx >= NB * COUT) return;
    int n  = idx >> 8;
    int co = idx & 255;
    const float4* cr = reinterpret_cast<const float4*>(c  + n  * CDIM);
    const float4* wr = reinterpret_cast<const float4*>(cw + co * CDIM);
    float s = bias[co];
#pragma unroll
    for (int i = 0; i < CDIM / 4; ++i) {
        float4 a = cr[i];
        float4 b = wr[i];
        s += a.x * b.x + a.y * b.y + a.z * b.z + a.w * b.w;
    }
    ctx[idx] = s;
}

// ---------------------------------------------------------------------------
// Kernel 2: weight transpose [co][ci][tap] -> wt[co][tap][ci]
// ---------------------------------------------------------------------------
__global__ void wtr_kernel(const float* __restrict__ w, float* __restrict__ wt) {
    int o = blockIdx.x * blockDim.x + threadIdx.x;
    if (o >= COUT * KTOT) return;
    int ci  = o & 31;
    int tap = (o >> 5) % 9;
    int co  = o / KTOT;
    wt[o] = w[(co * CING + ci) * 9 + tap];
}

// ---------------------------------------------------------------------------
// Kernel 3: grouped conv as implicit GEMM, async-LDS-staged, tap-major K.
// Block = 64 co (one group) x 224 pos (4 rows x 56 cols), 8 waves.
// Wave  = 16 co x 112 pos (7 x 16x16 C tiles), K = 9 taps x 32 ci.
// Steady state per WMMA: 1 ds_load_b64 (B) + 1/7 global float2 (A).
// ---------------------------------------------------------------------------
__global__ __launch_bounds__(256) void conv_wmma_kernel(
        const float* __restrict__ x,
        const float* __restrict__ wt,
        const float* __restrict__ ctx,
        float* __restrict__ out) {
    __shared__ float sX[8 * XBLK];               // 44544 B

    const int bid  = blockIdx.x;
    const int g    = bid & 3;                    // group
    const int rowt = (bid >> 2) % 14;            // 4-row tile
    const int n    = bid / 56;                   // batch
    const int h_base = rowt * 4;

    const int tid   = threadIdx.x;
    const int lane  = tid & 31;
    const int wid   = tid >> 5;
    const int laneM = lane & 15;
    const int kHi   = (lane >> 4) << 1;          // K sub-row 0 or 2
    const int co_t  = wid & 3;                   // wave's 16-co tile in group
    const int ph    = wid >> 2;                  // wave's 112-pos half

    const float* xb = x + (size_t)(n * CIN + g * CING) * HW;

    // ---- stage x tile (K-interleaved): async global->LDS for interior,
    // ds_store 0 for halo. Disjoint writers -> one asynccnt wait + barrier.
    for (int i = tid; i < CING * XROW; i += 256) {
        int ci  = i / XROW;
        int rem = i - ci * XROW;                 // row*58 + col
        int r   = rem / TCOLS;
        int cc  = rem - r * TCOLS;
        int gh  = h_base - 1 + r;
        int gw  = cc - 1;
        float* dst = &sX[(ci >> 2) * XBLK + rem * 4 + (ci & 3)];
        if (gh >= 0 && gh < HDIM && gw >= 0 && gw < WDIM)
            async_g2l_b32(dst, xb + ci * HW + gh * WDIM + gw);
        else
            *dst = 0.0f;
    }
    wait_async0();
    __syncthreads();

    // ---- per-lane position decode for the 7 sub-tiles ----
    int lrj[7], lcj[7];
#pragma unroll
    for (int j = 0; j < 7; ++j) {
        int lp = ph * 112 + j * 16 + laneM;      // 0..223 within block tile
        lrj[j] = lp / WDIM;                      // local row 0..3
        lcj[j] = lp % WDIM;                      // local col 0..55
    }

    const int co_row = g * COG + co_t * 16 + laneM;   // A-matrix row (M)
    const float* wrow = wt + (size_t)co_row * KTOT;

    v8f acc[7];
#pragma unroll
    for (int j = 0; j < 7; ++j) acc[j] = (v8f){};

#pragma unroll
    for (int tap = 0; tap < 9; ++tap) {
        const int kh = tap / 3;                  // constants after unroll
        const int kw = tap % 3;
        int boff[7];
#pragma unroll
        for (int j = 0; j < 7; ++j)
            boff[j] = ((lrj[j] + kh) * TCOLS + (lcj[j] + kw)) * 4;

        const float* wtap = wrow + tap * CING;

#pragma unroll 2
        for (int cib = 0; cib < 8; ++cib) {      // 4 K-rows per iteration
            // A fragment: two consecutive k weights, contiguous float2
            v2f a = *reinterpret_cast<const v2f*>(wtap + cib * 4 + kHi);
            // B base for this 4-ci block, this lane-half's K pair
            const float* s0 = sX + cib * XBLK + kHi;
#pragma unroll
            for (int j = 0; j < 7; ++j) {
                v2f b = *reinterpret_cast<const v2f*>(s0 + boff[j]);  // b64
                acc[j] = __builtin_amdgcn_wmma_f32_16x16x4_f32(
                             false, a, false, b, (short)0, acc[j], false, false);
            }
        }
    }

    // ---- epilogue: add ctx+bias, store ----
    const int mofs  = (lane >> 4) << 3;          // C-row offset 0 or 8
    const float* cb = ctx + n * COUT;
    float* ob       = out + (size_t)(n * COUT) * HW;
#pragma unroll
    for (int j = 0; j < 7; ++j) {
        const int pg = (h_base + lrj[j]) * WDIM + lcj[j];
#pragma unroll
        for (int r = 0; r < 8; ++r) {
            const int co = g * COG + co_t * 16 + mofs + r;
            ob[(size_t)co * HW + pg] = acc[j][r] + cb[co];
        }
    }
}

// ---------------------------------------------------------------------------
// Launch
// inputs: 0=x [32,128,56,56], 1=c [32,64], 2=weight [256,32,3,3],
//         3=bias [256], 4=c_weight [256,64]; out: [32,256,56,56] f32
// ws: [0, 8192) ctx table, [8192, 81920) transposed weights (320 KB)
// ---------------------------------------------------------------------------
extern "C" void kernel_launch(void* const* d_in, const int* in_sizes, int n_in,
                              void* d_out, int out_size, void* d_ws, size_t ws_size,
                              hipStream_t stream) {
    const float* x    = (const float*)d_in[0];
    const float* c    = (const float*)d_in[1];
    const float* wgt  = (const float*)d_in[2];
    const float* bias = (const float*)d_in[3];
    const float* cwgt = (const float*)d_in[4];
    float*       out  = (float*)d_out;
    float*       wsf  = (float*)d_ws;
    float*       ctx  = wsf;                       // 8192 floats
    float*       wt   = wsf + NB * COUT;           // 73728 floats

    ctx_bias_kernel<<<(NB * COUT + 255) / 256, 256, 0, stream>>>(c, cwgt, bias, ctx);
    wtr_kernel<<<(COUT * KTOT + 255) / 256, 256, 0, stream>>>(wgt, wt);

    // 32 n * 14 row-tiles * 4 groups = 1792 blocks of 8 waves
    conv_wmma_kernel<<<NB * 14 * GROUPS, 256, 0, stream>>>(x, wt, ctx, out);
}